// LinearAggregatedAttention_8160437862499
// MI455X (gfx1250) — compile-verified
//
#include <hip/hip_runtime.h>
#include <math.h>

// ---------------- problem constants ----------------
#define B_     8
#define H_     56
#define W_     56
#define C_     128
#define NH_    4
#define N_PIX  3136      // H*W
#define D_     32        // C/NH
#define L_     9         // WIN*WIN
#define P_     49        // PH*PW
#define PH_    7
#define PW_    7
#define TWO_C  256
#define NCAT   400       // 256 (kv) + 128 (sr) + 4 (q) + 4 (csum) + 8 pad
__device__ __constant__ float SCALE_ = 0.08838834764831845f; // C^-0.5

// ---------------- WMMA types ----------------
typedef __attribute__((ext_vector_type(2))) float v2f;
typedef __attribute__((ext_vector_type(8))) float v8f;

__device__ __forceinline__ float gelu_exact(float v) {
    return 0.5f * v * (1.0f + erff(v * 0.70710678118654752f));
}

// =====================================================================
// fp32 WMMA GEMM, K fixed at 128:  acc = A[M,128] @ W[Nout,128]^T
// One 16x16 tile per wave, 8 waves/block. Inner loop: 4 chunks of
// 8 K-steps; 16 fragment loads issued per chunk before the 8 WMMAs so
// VMEM latency overlaps the matrix pipe (no per-WMMA s_wait_loadcnt 0).
//
// MODE 0: Y0 = acc + bias
// MODE 1: Y0 = gelu(acc + bias)
// MODE 2: fused quad-output over the packed Wcat (NCAT cols):
//         col<256            -> kv   (B*N, 256)
//         256<=col<384       -> xs   (B*N, 128)   + gelu
//         384<=col<388       -> qb   (B, NH, N)   transposed store
//         388<=col<392       -> csum (B*N, NH)
//         col>=392           -> dropped (zero-pad columns)
// APERM: A viewed as (B*N, C) but stored (B, NH, N, D) (proj GEMM).
// =====================================================================
template<int MODE, bool APERM>
__global__ void gemm_wmma_f32(const float* __restrict__ A,
                              const float* __restrict__ Wt,
                              const float* __restrict__ bias,
                              float* __restrict__ Y0,
                              float* __restrict__ Y1,
                              float* __restrict__ Y2,
                              float* __restrict__ Y3,
                              int M, int Nout, int Npix)
{
    constexpr int KK = 128;
    const int lane   = threadIdx.x & 31;
    const int wave   = threadIdx.x >> 5;
    const int ntiles = Nout >> 4;
    const int mtiles = (M + 15) >> 4;
    const int t      = blockIdx.x * 8 + wave;
    if (t >= mtiles * ntiles) return;           // wave-uniform exit

    const int mt = t / ntiles, nt = t % ntiles;
    const int m0 = mt << 4,    n0 = nt << 4;

    const int mrow = lane & 15;
    const int kk   = (lane >> 4) << 1;          // 0 or 2 (K sub-slot)
    int arow = m0 + mrow; if (arow >= M) arow = M - 1;   // clamped read row
    const int bcol = n0 + (lane & 15);

    // A base address (APERM: c = h*32+d  ->  (b, h, n, d))
    const float* Abase;
    if (!APERM) {
        Abase = A + (size_t)arow * KK;
    } else {
        const int bb = arow / Npix, n = arow - bb * Npix;
        Abase = A + ((size_t)bb * NH_) * Npix * D_ + (size_t)n * D_;
    }
    const float* Bbase = Wt + (size_t)bcol * KK;

    v8f acc = {};
#pragma unroll
    for (int kb = 0; kb < KK; kb += 32) {
        v2f areg[8], breg[8];
        if (!APERM)
            __builtin_prefetch(Abase + kb + 32, 0, 3);   // near-scope prefetch
#pragma unroll
        for (int j = 0; j < 8; ++j) {
            const int k = kb + 4 * j + kk;
            float2 av;
            if (!APERM) {
                av = *(const float2*)(Abase + k);
            } else {
                av = *(const float2*)(Abase + (size_t)(k >> 5) * Npix * D_ + (k & 31));
            }
            const float2 bv = *(const float2*)(Bbase + k);
            areg[j].x = av.x; areg[j].y = av.y;
            breg[j].x = bv.x; breg[j].y = bv.y;
        }
#pragma unroll
        for (int j = 0; j < 8; ++j)
            acc = __builtin_amdgcn_wmma_f32_16x16x4_f32(
                      false, areg[j], false, breg[j], (short)0, acc, false, false);
    }

    const int   col  = n0 + (lane & 15);
    const float bval = bias[col];
    const int   rhi  = (lane >> 4) << 3;        // 0 or 8
#pragma unroll
    for (int r = 0; r < 8; ++r) {
        const int row = m0 + r + rhi;
        if (row >= M) continue;
        float v = acc[r] + bval;
        if (MODE == 0) {
            Y0[(size_t)row * Nout + col] = v;
        } else if (MODE == 1) {
            Y0[(size_t)row * Nout + col] = gelu_exact(v);
        } else {
            if (col < TWO_C) {
                Y0[(size_t)row * TWO_C + col] = v;                 // kv
            } else if (col < TWO_C + C_) {
                Y1[(size_t)row * C_ + (col - TWO_C)] = gelu_exact(v); // xs
            } else if (col < TWO_C + C_ + NH_) {
                const int bb = row / N_PIX, n = row - bb * N_PIX;
                const int h  = col - (TWO_C + C_);
                Y2[((size_t)bb * NH_ + h) * N_PIX + n] = v;        // qb (B,NH,N)
            } else if (col < TWO_C + C_ + 2 * NH_) {
                Y3[(size_t)row * NH_ + (col - (TWO_C + C_ + NH_))] = v; // csum
            }
        }
    }
}

// =====================================================================
// Pack Wcat (400x128) / bcat (400):
//   rows   0..255 : kv_w              bias kv_b
//   rows 256..383 : sr_w              bias sr_b
//   rows 384..387 : q_w               bias q_b
//   rows 388..391 : sum_d kv_w[h*32+d]  bias sum_d kv_b[h*32+d]
//   rows 392..399 : zeros
// =====================================================================
__global__ void pack_wcat_kernel(const float* __restrict__ kv_w,
                                 const float* __restrict__ kv_b,
                                 const float* __restrict__ sr_w,
                                 const float* __restrict__ sr_b,
                                 const float* __restrict__ q_w,
                                 const float* __restrict__ q_b,
                                 float* __restrict__ Wcat,
                                 float* __restrict__ bcat)
{
    const int i = blockIdx.x * 256 + threadIdx.x;   // row*128 + k
    if (i >= NCAT * C_) return;
    const int row = i >> 7, k = i & 127;
    float v;
    if (row < TWO_C)            v = kv_w[i];
    else if (row < TWO_C + C_)  v = sr_w[(row - TWO_C) * C_ + k];
    else if (row < TWO_C + C_ + NH_) v = q_w[(row - (TWO_C + C_)) * C_ + k];
    else if (row < TWO_C + C_ + 2 * NH_) {
        const int g = row - (TWO_C + C_ + NH_);
        float s = 0.f;
        for (int d = 0; d < D_; ++d) s += kv_w[(g * D_ + d) * C_ + k];
        v = s;
    } else v = 0.f;
    Wcat[i] = v;

    if (k == 0) {
        float bv;
        if (row < TWO_C)            bv = kv_b[row];
        else if (row < TWO_C + C_)  bv = sr_b[row - TWO_C];
        else if (row < TWO_C + C_ + NH_) bv = q_b[row - (TWO_C + C_)];
        else if (row < TWO_C + C_ + 2 * NH_) {
            const int g = row - (TWO_C + C_ + NH_);
            float s = 0.f;
            for (int d = 0; d < D_; ++d) s += kv_b[g * D_ + d];
            bv = s;
        } else bv = 0.f;
        bcat[row] = bv;
    }
}

// =====================================================================
// context_score: per-(b,h) softmax over N of (q + query_embedding[h])
// =====================================================================
__global__ void softmax_q_kernel(const float* __restrict__ qb,
                                 const float* __restrict__ qemb,
                                 float* __restrict__ cs)
{
    const int bh = blockIdx.x;
    const int h  = bh & (NH_ - 1);
    const float qe = qemb[h];
    const float* src = qb + (size_t)bh * N_PIX;
    float* dst = cs + (size_t)bh * N_PIX;

    __shared__ float red[256];
    const int tid = threadIdx.x;

    float m = -3.4e38f;
    for (int i = tid; i < N_PIX; i += 256) m = fmaxf(m, src[i] + qe);
    red[tid] = m; __syncthreads();
    for (int s = 128; s > 0; s >>= 1) {
        if (tid < s) red[tid] = fmaxf(red[tid], red[tid + s]);
        __syncthreads();
    }
    m = red[0]; __syncthreads();

    float sum = 0.f;
    for (int i = tid; i < N_PIX; i += 256) sum += expf(src[i] + qe - m);
    red[tid] = sum; __syncthreads();
    for (int s = 128; s > 0; s >>= 1) {
        if (tid < s) red[tid] += red[tid + s];
        __syncthreads();
    }
    const float inv = 1.0f / red[0];

    for (int i = tid; i < N_PIX; i += 256)
        dst[i] = expf(src[i] + qe - m) * inv;
}

// =====================================================================
// 8x8 average pool of xs (stored (B,N,C)) + LayerNorm over C.
// =====================================================================
__global__ void pool_ln_kernel(const float* __restrict__ xs,
                               const float* __restrict__ nw,
                               const float* __restrict__ nb,
                               float* __restrict__ xpn)
{
    const int bp = blockIdx.x;
    const int b  = bp / P_, p = bp - b * P_;
    const int ph = p / PW_, pw = p - ph * PW_;
    const int c  = threadIdx.x;

    float s = 0.f;
#pragma unroll 8
    for (int k = 0; k < 64; ++k) {
        const int yy = ph * 8 + (k >> 3), xx = pw * 8 + (k & 7);
        s += xs[((size_t)b * N_PIX + yy * W_ + xx) * C_ + c];
    }
    const float v = s * (1.0f / 64.0f);

    __shared__ float r1[128], r2[128];
    r1[c] = v; r2[c] = v * v; __syncthreads();
    for (int st = 64; st > 0; st >>= 1) {
        if (c < st) { r1[c] += r1[c + st]; r2[c] += r2[c + st]; }
        __syncthreads();
    }
    const float mu  = r1[0] * (1.0f / 128.0f);
    const float var = r2[0] * (1.0f / 128.0f) - mu * mu;
    xpn[(size_t)bp * C_ + c] = (v - mu) * rsqrtf(var + 1e-5f) * nw[c] + nb[c];
}

// k_pool[b,h,p] = sum_d kvp[b,p, h*32+d]
__global__ void kpool_kernel(const float* __restrict__ kvp,
                             float* __restrict__ kp)
{
    const int i = blockIdx.x * 256 + threadIdx.x;
    if (i >= B_ * NH_ * P_) return;
    const int p = i % P_, bh = i / P_;
    const int h = bh & (NH_ - 1), b = bh >> 2;
    const float* r = kvp + ((size_t)b * P_ + p) * TWO_C + h * D_;
    float s = 0.f;
#pragma unroll
    for (int d = 0; d < D_; ++d) s += r[d];
    kp[i] = s;
}

// =====================================================================
// Pass 1: per (b,h,n): attn[local(9)|pool(49)], softmax(58),
// x_pool = a_pool @ v_pool (v_pool in LDS), deterministic block
// partials of sum_n a_local.
// =====================================================================
__global__ void attn_pass1_kernel(const float* __restrict__ cs_buf,
                                  const float* __restrict__ csum,
                                  const float* __restrict__ kvp,
                                  const float* __restrict__ kpool,
                                  const float* __restrict__ pbp,   // (NH,N,P)
                                  const float* __restrict__ pbl,   // (NH,L)
                                  float* __restrict__ xout,        // (B,NH,N,D)
                                  float* __restrict__ wpart)       // (B*NH, grid.x, L)
{
    const int bh = blockIdx.y;
    const int h  = bh & (NH_ - 1), b = bh >> 2;
    const int n  = blockIdx.x * 256 + threadIdx.x;
    const bool active = (n < N_PIX);

    __shared__ float vp[P_ * D_];
    __shared__ float kp[P_];
    __shared__ float red[256];

    for (int i = threadIdx.x; i < P_ * D_; i += 256) {
        const int p = i >> 5, d = i & 31;
        vp[i] = kvp[((size_t)b * P_ + p) * TWO_C + C_ + h * D_ + d];
    }
    for (int i = threadIdx.x; i < P_; i += 256)
        kp[i] = kpool[(size_t)bh * P_ + i];
    __syncthreads();

    float al[L_];
    float ap[P_];

    if (active) {
        const float cs = cs_buf[(size_t)bh * N_PIX + n];
        const int y = n / W_, x = n - y * W_;
        float mx = -3.4e38f;
#pragma unroll
        for (int l = 0; l < L_; ++l) {
            const int dy = l / 3 - 1, dx = l % 3 - 1;
            const int yy = y + dy, xx = x + dx;
            const bool valid = ((unsigned)yy < H_) && ((unsigned)xx < W_);
            const float ks = valid
                ? csum[((size_t)b * N_PIX + yy * W_ + xx) * NH_ + h] : 0.f;
            al[l] = valid ? fmaf(ks * cs, SCALE_, pbl[h * L_ + l]) : -INFINITY;
            mx = fmaxf(mx, al[l]);
        }
        const float* pb = pbp + ((size_t)h * N_PIX + n) * P_;
#pragma unroll
        for (int p = 0; p < P_; ++p) {
            ap[p] = fmaf(cs * kp[p], SCALE_, pb[p]);
            mx = fmaxf(mx, ap[p]);
        }
        float ssum = 0.f;
#pragma unroll
        for (int l = 0; l < L_; ++l) { al[l] = expf(al[l] - mx); ssum += al[l]; }
#pragma unroll
        for (int p = 0; p < P_; ++p) { ap[p] = expf(ap[p] - mx); ssum += ap[p]; }
        const float inv = 1.0f / ssum;
#pragma unroll
        for (int l = 0; l < L_; ++l) al[l] *= inv;

        float accd[D_];
#pragma unroll
        for (int d = 0; d < D_; ++d) accd[d] = 0.f;
#pragma unroll
        for (int p = 0; p < P_; ++p) {
            const float a = ap[p] * inv;
            const float* vr = &vp[p * D_];
#pragma unroll
            for (int d = 0; d < D_; ++d) accd[d] = fmaf(a, vr[d], accd[d]);
        }
        float* o = xout + ((size_t)bh * N_PIX + n) * D_;
#pragma unroll
        for (int d = 0; d < D_; ++d) o[d] = accd[d];
    }

    // deterministic per-block tree reduction of a_local over the 256 rows
#pragma unroll
    for (int l = 0; l < L_; ++l) {
        __syncthreads();
        red[threadIdx.x] = active ? al[l] : 0.f;
        __syncthreads();
        for (int s = 128; s > 0; s >>= 1) {
            if (threadIdx.x < s) red[threadIdx.x] += red[threadIdx.x + s];
            __syncthreads();
        }
        if (threadIdx.x == 0)
            wpart[((size_t)bh * gridDim.x + blockIdx.x) * L_ + l] = red[0];
    }
}

// fixed-order sum of per-block partials -> wloc
__global__ void wloc_reduce_kernel(const float* __restrict__ wpart,
                                   float* __restrict__ wloc, int nblk)
{
    const int i = threadIdx.x;
    if (i >= B_ * NH_ * L_) return;
    const int bh = i / L_, l = i % L_;
    float s = 0.f;
    for (int k = 0; k < nblk; ++k)
        s += wpart[((size_t)bh * nblk + k) * L_ + l];
    wloc[i] = s;
}

// =====================================================================
// Pass 2: x_local gather-accumulate into xout (kv is L2-resident).
// w_local[l] = learnable_tokens + N*learnable_bias + sum_n a_local
// =====================================================================
__global__ void pass2_kernel(const float* __restrict__ kv,
                             const float* __restrict__ wloc,
                             const float* __restrict__ lt,
                             const float* __restrict__ lb,
                             float* __restrict__ xout)
{
    const int bh = blockIdx.y;
    const int h  = bh & (NH_ - 1), b = bh >> 2;
    const int n  = blockIdx.x * 256 + threadIdx.x;
    if (n >= N_PIX) return;

    float wl[L_];
#pragma unroll
    for (int l = 0; l < L_; ++l)
        wl[l] = lt[h * L_ + l] + (float)N_PIX * lb[h * L_ + l]
              + wloc[bh * L_ + l];

    const int y = n / W_, x = n - y * W_;
    float acc[D_];
#pragma unroll
    for (int d = 0; d < D_; ++d) acc[d] = 0.f;

#pragma unroll
    for (int l = 0; l < L_; ++l) {
        const int dy = l / 3 - 1, dx = l % 3 - 1;
        const int yy = y + dy, xx = x + dx;
        if (((unsigned)yy < H_) && ((unsigned)xx < W_)) {
            const float* v = kv + ((size_t)b * N_PIX + yy * W_ + xx) * TWO_C
                           + C_ + h * D_;
            const float w = wl[l];
#pragma unroll
            for (int d = 0; d < D_; ++d)
                acc[d] = fmaf(fmaxf(v[d], 0.f), w, acc[d]);
        }
    }
    float* o = xout + ((size_t)bh * N_PIX + n) * D_;
#pragma unroll
    for (int d = 0; d < D_; ++d) o[d] += acc[d];
}

// =====================================================================
// host-side launch
// =====================================================================
static inline size_t alignup(size_t v) { return (v + 255) & ~(size_t)255; }

extern "C" void kernel_launch(void* const* d_in, const int* in_sizes, int n_in,
                              void* d_out, int out_size, void* d_ws, size_t ws_size,
                              hipStream_t stream)
{
    const float* x     = (const float*)d_in[0];
    const float* q_w   = (const float*)d_in[1];
    const float* q_b   = (const float*)d_in[2];
    const float* qemb  = (const float*)d_in[3];
    const float* kv_w  = (const float*)d_in[4];
    const float* kv_b  = (const float*)d_in[5];
    const float* sr_w  = (const float*)d_in[6];
    const float* sr_b  = (const float*)d_in[7];
    const float* nw    = (const float*)d_in[8];
    const float* nb    = (const float*)d_in[9];
    const float* pbp   = (const float*)d_in[10];
    const float* pbl   = (const float*)d_in[11];
    const float* lt    = (const float*)d_in[12];
    const float* lb    = (const float*)d_in[13];
    const float* pj_w  = (const float*)d_in[14];
    const float* pj_b  = (const float*)d_in[15];
    float* out = (float*)d_out;

    // workspace carve-up
    char* base = (char*)d_ws;
    size_t off = 0;
    float* kv    = (float*)(base + off); off += alignup((size_t)B_*N_PIX*TWO_C*4);
    float* xs    = (float*)(base + off); off += alignup((size_t)B_*N_PIX*C_*4);
    float* xout  = (float*)(base + off); off += alignup((size_t)B_*N_PIX*C_*4);
    float* qb    = (float*)(base + off); off += alignup((size_t)B_*NH_*N_PIX*4);
    float* cs    = (float*)(base + off); off += alignup((size_t)B_*NH_*N_PIX*4);
    float* csum  = (float*)(base + off); off += alignup((size_t)B_*N_PIX*NH_*4);
    float* xpn   = (float*)(base + off); off += alignup((size_t)B_*P_*C_*4);
    float* kvp   = (float*)(base + off); off += alignup((size_t)B_*P_*TWO_C*4);
    float* kp    = (float*)(base + off); off += alignup((size_t)B_*NH_*P_*4);
    float* wloc  = (float*)(base + off); off += alignup((size_t)B_*NH_*L_*4);
    const int NBLK = (N_PIX + 255) / 256;                    // 13
    float* wpart = (float*)(base + off); off += alignup((size_t)B_*NH_*NBLK*L_*4);
    float* Wcat  = (float*)(base + off); off += alignup((size_t)NCAT*C_*4);
    float* bcat  = (float*)(base + off); off += alignup((size_t)NCAT*4);
    (void)ws_size; (void)n_in; (void)in_sizes; (void)out_size;

    const int M  = B_ * N_PIX;                               // 25088
    const int MT = (M + 15) / 16;                            // 1568

    // 0) pack concatenated weights (kv | sr | q | group-summed kv | pad)
    pack_wcat_kernel<<<(NCAT * C_ + 255) / 256, 256, 0, stream>>>(
        kv_w, kv_b, sr_w, sr_b, q_w, q_b, Wcat, bcat);

    // 1) fused front-end GEMM: single pass over x produces kv, xs(gelu),
    //    qb (transposed) and csum              (WMMA f32, Nout=400)
    gemm_wmma_f32<2, false><<<(MT * (NCAT/16) + 7) / 8, 256, 0, stream>>>(
        x, Wcat, bcat, kv, xs, qb, csum, M, NCAT, N_PIX);

    // 2) context_score softmax over N per (b,h)
    softmax_q_kernel<<<B_ * NH_, 256, 0, stream>>>(qb, qemb, cs);
    // 3) 8x8 avg pool + LayerNorm
    pool_ln_kernel<<<B_ * P_, 128, 0, stream>>>(xs, nw, nb, xpn);
    // 4) kvp = xpn @ kv_w^T + kv_b    (WMMA f32, M=392 with row guards)
    {
        const int Mp = B_ * P_, MTp = (Mp + 15) / 16;
        gemm_wmma_f32<0, false><<<(MTp * (TWO_C/16) + 7) / 8, 256, 0, stream>>>(
            xpn, kv_w, kv_b, kvp, nullptr, nullptr, nullptr, Mp, TWO_C, N_PIX);
    }
    // 5) k_pool
    kpool_kernel<<<(B_*NH_*P_ + 255) / 256, 256, 0, stream>>>(kvp, kp);
    // 6) fused attention pass 1
    attn_pass1_kernel<<<dim3(NBLK, B_ * NH_), 256, 0, stream>>>(
        cs, csum, kvp, kp, pbp, pbl, xout, wpart);
    // 7) deterministic w_local reduction
    wloc_reduce_kernel<<<1, 320, 0, stream>>>(wpart, wloc, NBLK);
    // 8) pass 2: x_local gather-accumulate into xout
    pass2_kernel<<<dim3(NBLK, B_ * NH_), 256, 0, stream>>>(kv, wloc, lt, lb, xout);
    // 9) out = xout(permuted view) @ proj_w^T + proj_b   (WMMA f32)
    gemm_wmma_f32<0, true><<<(MT * (C_/16) + 7) / 8, 256, 0, stream>>>(
        xout, pj_w, pj_b, out, nullptr, nullptr, nullptr, M, C_, N_PIX);
}